// PricePredictionLoss_67534065762422
// MI455X (gfx1250) — compile-verified
//
#include <hip/hip_runtime.h>
#include <stdint.h>

// ---------------------------------------------------------------------------
// PricePredictionLoss on MI455X (gfx1250, wave32)
//
// 80MB in -> 1 float out; roughly balanced between HBM (3.4us @ 23.3TB/s)
// and VALU (~7e8 f32 ops). CDNA5 path: 40B rows are staged into LDS with the
// async copy engine (global_load_async_to_lds_b128, ASYNCcnt), DOUBLE
// BUFFERED with a partial wait (s_wait_asynccnt 6) so tile n+1's HBM traffic
// overlaps tile n's compute. Each wave issues exactly 6 async ops per tile
// (uniform EXEC!=0 on every issue) so the partial-wait count is exact.
// WMMA intentionally unused: the only matmul view is [B,10]x[10,45] diff
// matrices (K=10) with a heavy per-element epilogue -- VALU wins.
// ---------------------------------------------------------------------------

#define SEQ        10
#define NPAIR      45
#define TILE_ROWS  256
#define TILE_FLT   (TILE_ROWS * SEQ)   // 2560 floats per array per tile
#define TILE_V4    (TILE_FLT / 4)      // 640 float4 chunks
#define V4_PER_WV  (TILE_V4 / 8)       // 80 chunks per wave (8 waves/block)
#define NACC       (3 + 2 * NPAIR)     // 93 global accumulators
#define THRESH_C   0.01f
#define EPS_C      1e-8f

// ---- CDNA5 async global->LDS copy (ASYNCcnt), via inline asm -------------
__device__ __forceinline__ unsigned lds_off(const void* shared_ptr) {
  // flat shared address: low 32 bits are the LDS byte offset (HW truncates)
  return (unsigned)(uintptr_t)shared_ptr;
}
__device__ __forceinline__ void async_ld_b128(unsigned dst_lds_off, const void* gptr) {
  asm volatile("global_load_async_to_lds_b128 %0, %1, off"
               :: "v"(dst_lds_off), "v"((unsigned long long)(uintptr_t)gptr)
               : "memory");
}
__device__ __forceinline__ void wait_async_le6() {   // prev tile done, next in flight
  asm volatile("s_wait_asynccnt 6" ::: "memory");
}
__device__ __forceinline__ void wait_async_le0() {   // full drain
  asm volatile("s_wait_asynccnt 0" ::: "memory");
}

__device__ __forceinline__ float wave_sum32(float v) {
#pragma unroll
  for (int m = 16; m >= 1; m >>= 1) v += __shfl_xor(v, m, 32);
  return v;
}

__global__ __launch_bounds__(256) void loss_partial_kernel(
    const float* __restrict__ pred, const float* __restrict__ targ,
    const float* __restrict__ tw, float* __restrict__ ws,
    int B, int nTiles) {
  __shared__ __align__(16) float s_p[2][TILE_FLT];
  __shared__ __align__(16) float s_t[2][TILE_FLT];
  __shared__ float s_part[8][NACC + 1];  // +1 pad

  const int tid  = threadIdx.x;
  const int lane = tid & 31;
  const int wave = tid >> 5;
  const int cbase = wave * V4_PER_WV;

  // Issue one tile's copies: exactly 6 async ops per wave (3 issues x 2 arrays;
  // 3rd issue has lanes 0..15 active -> EXEC!=0 -> always counted).
  auto issue_tile = [&](int tile, int buf) {
    const float* gp = pred + (long long)tile * TILE_FLT;
    const float* gt = targ + (long long)tile * TILE_FLT;
#pragma unroll
    for (int it = 0; it < 3; ++it) {
      const int rel = lane + it * 32;
      if (rel < V4_PER_WV) {
        const int c = cbase + rel;
        async_ld_b128(lds_off(&s_p[buf][c * 4]), gp + c * 4);
        async_ld_b128(lds_off(&s_t[buf][c * 4]), gt + c * 4);
      }
    }
  };

  // temporal weights: uniform -> scalar loads, kept in registers
  float w[SEQ];
#pragma unroll
  for (int j = 0; j < SEQ; ++j) w[j] = tw[j];

  float base_acc = 0.f, dir_acc = 0.f, trend_acc = 0.f;
  float err_acc[NPAIR], cnt_acc[NPAIR];
#pragma unroll
  for (int k = 0; k < NPAIR; ++k) { err_acc[k] = 0.f; cnt_acc[k] = 0.f; }

  // ---- double-buffered pipeline over this block's tiles --------------------
  const int tile0 = blockIdx.x;
  if (tile0 < nTiles && (long long)(tile0 + 1) * TILE_ROWS <= (long long)B)
    issue_tile(tile0, 0);  // prime buffer 0

  int mbuf = 0;
  for (int tile = tile0; tile < nTiles; tile += gridDim.x, mbuf ^= 1) {
    const long long rowBase = (long long)tile * TILE_ROWS;
    const bool full = (rowBase + TILE_ROWS) <= (long long)B;       // uniform
    const int  nextTile = tile + gridDim.x;
    const bool nextFull = (nextTile < nTiles) &&
        ((long long)(nextTile + 1) * TILE_ROWS <= (long long)B);   // uniform

    if (nextFull) issue_tile(nextTile, mbuf ^ 1);  // prefetch next tile

    float p[SEQ], t[SEQ];
    bool haveRow = true;
    if (full) {
      if (nextFull) wait_async_le6();  // this tile done; next still in flight
      else          wait_async_le0();
      __syncthreads();                 // all waves' copies visible
#pragma unroll
      for (int j = 0; j < SEQ; ++j) {
        p[j] = s_p[mbuf][tid * SEQ + j];
        t[j] = s_t[mbuf][tid * SEQ + j];
      }
    } else {
      const long long row = rowBase + tid;
      haveRow = row < (long long)B;
      if (haveRow) {
#pragma unroll
        for (int j = 0; j < SEQ; ++j) {
          p[j] = pred[row * SEQ + j];
          t[j] = targ[row * SEQ + j];
        }
      }
    }

    if (haveRow) {
      // 1) temporally weighted squared error
#pragma unroll
      for (int j = 0; j < SEQ; ++j) {
        float d = p[j] - t[j];
        base_acc = fmaf(d * d, w[j], base_acc);
      }
      // 2) direction loss
#pragma unroll
      for (int j = 0; j < SEQ - 1; ++j) {
        float pc = p[j + 1] - p[j];
        float tc = t[j + 1] - t[j];
        bool mm = ((pc > 0.f) != (tc > 0.f)) || ((pc < 0.f) != (tc < 0.f));
        dir_acc += mm ? (1.f + fabsf(tc)) : 0.f;
      }
      // 3) trend loss (second differences)
#pragma unroll
      for (int j = 0; j < SEQ - 2; ++j) {
        float pt = p[j + 2] - 2.f * p[j + 1] + p[j];
        float tt = t[j + 2] - 2.f * t[j + 1] + t[j];
        float d = pt - tt;
        trend_acc = fmaf(d, d, trend_acc);
      }
      // 4) pairwise ranking: 45 independent masked chains (ILP)
      int k = 0;
#pragma unroll
      for (int i = 0; i < SEQ - 1; ++i) {
#pragma unroll
        for (int j = i + 1; j < SEQ; ++j) {
          float td = t[i] - t[j];
          float pd = p[i] - p[j];
          float atd = fabsf(td);
          float r = __builtin_amdgcn_rcpf(atd + EPS_C);   // v_rcp_f32
          float err = fmaxf(0.f, fmaf(-(td * pd), r, 1.f));
          float s = (atd > THRESH_C) ? 1.f : 0.f;
          err_acc[k] = fmaf(s, err, err_acc[k]);
          cnt_acc[k] += s;
          ++k;
        }
      }
    }

    if (full) __syncthreads();  // everyone done reading buf[mbuf]; it may be
                                // overwritten by the issue two tiles later
  }

  // ---- wave32 reduction, block combine, 93 global f32 atomics --------------
  base_acc  = wave_sum32(base_acc);
  dir_acc   = wave_sum32(dir_acc);
  trend_acc = wave_sum32(trend_acc);
#pragma unroll
  for (int k = 0; k < NPAIR; ++k) {
    err_acc[k] = wave_sum32(err_acc[k]);
    cnt_acc[k] = wave_sum32(cnt_acc[k]);
  }

  if (lane == 0) {
    s_part[wave][0] = base_acc;
    s_part[wave][1] = dir_acc;
    s_part[wave][2] = trend_acc;
#pragma unroll
    for (int k = 0; k < NPAIR; ++k) {
      s_part[wave][3 + k]         = err_acc[k];
      s_part[wave][3 + NPAIR + k] = cnt_acc[k];
    }
  }
  __syncthreads();
  if (tid < NACC) {
    float s = 0.f;
#pragma unroll
    for (int wv = 0; wv < 8; ++wv) s += s_part[wv][tid];
    atomicAdd(&ws[tid], s);   // global_atomic_add_f32, fire-and-forget
  }
}

__global__ void loss_final_kernel(const float* __restrict__ ws,
                                  float* __restrict__ out, int B) {
  if (threadIdx.x == 0 && blockIdx.x == 0) {
    float base  = ws[0] / (float)(B * SEQ);
    float dir   = ws[1] / (float)(B * (SEQ - 1));
    float trend = ws[2] / (float)(B * (SEQ - 2));
    float sum = 0.f, npairs = 0.f;
    for (int k = 0; k < NPAIR; ++k) {
      float c = ws[3 + NPAIR + k];
      if (c > 0.f) {
        sum += ws[3 + k] / fmaxf(c, 1.f);
        npairs += 1.f;
      }
    }
    float ranking = sum / fmaxf(npairs, 1.f);
    out[0] = base + 0.3f * dir + 0.2f * trend + 0.1f * ranking;
  }
}

extern "C" void kernel_launch(void* const* d_in, const int* in_sizes, int n_in,
                              void* d_out, int out_size, void* d_ws, size_t ws_size,
                              hipStream_t stream) {
  const float* pred = (const float*)d_in[0];
  const float* targ = (const float*)d_in[1];
  const float* tw   = (const float*)d_in[2];
  float* ws = (float*)d_ws;

  int B = in_sizes[0] / SEQ;
  int nTiles = (B + TILE_ROWS - 1) / TILE_ROWS;
  int grid = nTiles < 512 ? nTiles : 512;
  if (grid < 1) grid = 1;

  hipMemsetAsync(d_ws, 0, NACC * sizeof(float), stream);
  loss_partial_kernel<<<grid, 256, 0, stream>>>(pred, targ, tw, ws, B, nTiles);
  loss_final_kernel<<<1, 32, 0, stream>>>(ws, (float*)d_out, B);
}